// IDMForwardSim_76596446757215
// MI455X (gfx1250) — compile-verified
//
#include <hip/hip_runtime.h>
#include <math.h>

// ---------------------------------------------------------------------------
// IDM forward scan, B=262144 batches x T=40 sequential steps.
// Memory-bound (~425 MB @ 23.3 TB/s => ~18 us floor). One thread per batch,
// sequential carry over T. CDNA5 path: async global->LDS DMA
// (global_load_async_to_lds_b128, ASYNCcnt, s_wait_asynccnt) with per-wave
// double-buffered staging to fix the 1280B-per-thread stride of idm_s.
// Chunk loop fully unrolled: no branches, constant wait immediates.
// ---------------------------------------------------------------------------

#define T_STEPS 40
#define CT      4                   // timesteps per staged chunk
#define NCHUNK  (T_STEPS / CT)      // 10 chunks
#define NW      4                   // waves per block
#define BLOCK   (NW * 32)           // 128 threads
#define ROWF    36                  // floats per lane row in sbuf (32 data + 4 pad = 144B, 16B aligned)

// Async copy: 16B per lane; LDS dest addr VGPR, 32-bit global offset VGPR,
// 64-bit SGPR base. The same immediate offset applies to BOTH the global and
// LDS side (CDNA5 ISA GVS pseudocode) — exactly what we want since
// each LDS row mirrors the batch's 128B global segment.
#define ASYNC_CP_B128(LDS, GOFF, BASE, OFFSTR)                                 \
  asm volatile("global_load_async_to_lds_b128 %0, %1, %2 offset:" OFFSTR       \
               :: "v"(LDS), "v"(GOFF), "s"(BASE) : "memory")

__device__ __forceinline__ void issue_chunk(unsigned lds_s, int goff_s, const float* sbase,
                                            unsigned lds_a, int goff_a, const float* abase) {
  // protect address VGPRs of in-flight VMEM against XNACK replay re-reads
  asm volatile("s_wait_xcnt 0x0" ::: "memory");
  ASYNC_CP_B128(lds_s, goff_s, sbase, "0");
  ASYNC_CP_B128(lds_s, goff_s, sbase, "16");
  ASYNC_CP_B128(lds_s, goff_s, sbase, "32");
  ASYNC_CP_B128(lds_s, goff_s, sbase, "48");
  ASYNC_CP_B128(lds_s, goff_s, sbase, "64");
  ASYNC_CP_B128(lds_s, goff_s, sbase, "80");
  ASYNC_CP_B128(lds_s, goff_s, sbase, "96");
  ASYNC_CP_B128(lds_s, goff_s, sbase, "112");
  ASYNC_CP_B128(lds_a, goff_a, abase, "0");   // 9 async ops per chunk total
}

__global__ __launch_bounds__(BLOCK) void idm_scan_kernel(
    const float* __restrict__ att,     // [B, T]
    const float* __restrict__ params,  // [B, 5]
    const float* __restrict__ idm_s,   // [B, T, 8]
    float* __restrict__ out)           // [B, T] (trailing dim 1 flattened)
{
  // per-wave private double buffers (no cross-wave sharing -> no barriers)
  __shared__ float s_s[NW][2][32 * ROWF];  // 36864 B
  __shared__ float s_a[NW][2][32 * CT];    //  4096 B  => 40 KB/block, 8 blocks/WGP

  const int lane = threadIdx.x & 31;
  const int wave = threadIdx.x >> 5;
  const int b    = blockIdx.x * BLOCK + threadIdx.x;   // grid covers B exactly

  // Low 32 bits of a generic pointer to __shared__ == workgroup-relative LDS offset
  unsigned ldsS[2], ldsA[2];
  ldsS[0] = (unsigned)(size_t)&s_s[wave][0][lane * ROWF];
  ldsS[1] = (unsigned)(size_t)&s_s[wave][1][lane * ROWF];
  ldsA[0] = (unsigned)(size_t)&s_a[wave][0][lane * CT];
  ldsA[1] = (unsigned)(size_t)&s_a[wave][1][lane * CT];

  const int gs0 = b * (T_STEPS * 8 * 4);  // byte offset of this batch's idm_s row
  const int ga0 = b * (T_STEPS * 4);      // byte offset of this batch's att row

  // software pipeline: prefetch chunks 0 and 1
  issue_chunk(ldsS[0], gs0,                idm_s, ldsA[0], ga0,               att);
  issue_chunk(ldsS[1], gs0 + 1 * CT * 32,  idm_s, ldsA[1], ga0 + 1 * CT * 4,  att);

  // per-batch parameters (read once; cache lines shared by ~6 adjacent threads)
  const float* pp = params + b * 5;
  const float dvP = pp[0];   // desired_v
  const float tg  = pp[1];   // tgap
  const float jx  = pp[2];   // jamx
  const float ma  = pp[3];   // max_act
  const float mi  = pp[4];   // min_act
  const float inv_dv = 1.0f / dvP;
  const float inv2s  = 1.0f / (2.0f * sqrtf(ma * mi));

  // chunk 0 resident (18 in flight -> wait until only the newest 9 remain)
  asm volatile("s_wait_asynccnt 0x9" ::: "memory");
  float ego_v = s_s[wave][0][lane * ROWF + 0];
  float ego_x = s_s[wave][0][lane * ROWF + 3];

  float* outp = out + b * T_STEPS;

#pragma unroll
  for (int c = 0; c < NCHUNK; ++c) {
    // async loads complete in order: wait<=9 => chunk c fully in LDS while
    // chunk c+1's 9 ops stream in the background. Last chunk: drain to 0.
    // (constant-folds under full unroll -> fixed immediates, no branches)
    if (c + 1 < NCHUNK) { asm volatile("s_wait_asynccnt 0x9" ::: "memory"); }
    else                { asm volatile("s_wait_asynccnt 0x0" ::: "memory"); }

    const int buf = c & 1;
    const float* srow = &s_s[wave][buf][lane * ROWF];
    const float* arow = &s_a[wave][buf][lane * CT];

    float accv[CT];
#pragma unroll
    for (int tl = 0; tl < CT; ++tl) {
      const float4 w0 = *(const float4*)(srow + tl * 8);      // {ego_v, leader_v, merger_v, ego_x}
      const float4 w1 = *(const float4*)(srow + tl * 8 + 4);  // {leader_x, merger_x, le, me}
      const float a  = arow[tl];
      const float lv = w0.y, mv = w0.z;
      const float lx = w1.x, mx = w1.y, le = w1.z, me = w1.w;

      const float r    = ego_v * inv_dv;
      const float r2   = r * r;
      const float r4c  = 1.0f - r2 * r2;        // 1 - (v/v0)^4, shared by both branches
      const float tgv  = tg * ego_v;
      const float vter = ego_v * inv2s;

      // leader branch
      const float dvl  = (ego_v - lv) * le;
      const float dxl  = fmaf(le, (lx - ego_x) - 1000.0f, 1000.0f);
      const float gapl = jx + fmaxf(0.0f, fmaf(dvl, vter, tgv));
      const float ql   = gapl * __builtin_amdgcn_rcpf(dxl);
      float al = ma * (r4c - ql * ql);
      al = fminf(3.0f, fmaxf(-3.0f, al));

      // merger branch
      const float dvm  = (ego_v - mv) * me;
      const float dxm  = fmaf(me, (mx - ego_x) - 1000.0f, 1000.0f);
      const float gapm = jx + fmaxf(0.0f, fmaf(dvm, vter, tgv));
      const float qm   = gapm * __builtin_amdgcn_rcpf(dxm);
      float am = ma * (r4c - qm * qm);
      am = fminf(3.0f, fmaxf(-3.0f, am));

      const float act = al + a * (am - al);     // (1-a)*al + a*am
      ego_v = fmaf(act, 0.1f, ego_v);           // new_v = v + act*DT
      ego_x = ego_x + fmaf(act, 0.005f, ego_v * 0.1f);  // x + new_v*DT + 0.5*act*DT^2
      accv[tl] = act;
    }

    // one aligned 16B store per chunk per lane
    *(float4*)(outp + c * CT) = make_float4(accv[0], accv[1], accv[2], accv[3]);

    // refill this buffer with chunk c+2 (all reads above already consumed)
    if (c + 2 < NCHUNK) {
      issue_chunk(ldsS[buf], gs0 + (c + 2) * CT * 32, idm_s,
                  ldsA[buf], ga0 + (c + 2) * CT * 4,  att);
    }
  }
}

extern "C" void kernel_launch(void* const* d_in, const int* in_sizes, int n_in,
                              void* d_out, int out_size, void* d_ws, size_t ws_size,
                              hipStream_t stream) {
  const float* att    = (const float*)d_in[0];  // [B,T] f32
  const float* params = (const float*)d_in[1];  // [B,5] f32
  const float* s      = (const float*)d_in[2];  // [B,T,8] f32
  float* out          = (float*)d_out;          // [B,T,1] f32

  const int B = in_sizes[1] / 5;                // 262144 (multiple of BLOCK)
  const int grid = B / BLOCK;                   // 2048, exact cover
  idm_scan_kernel<<<grid, BLOCK, 0, stream>>>(att, params, s, out);
}